// PaoModel_73143293051497
// MI455X (gfx1250) — compile-verified
//
#include <hip/hip_runtime.h>
#include <math.h>

typedef float v2f __attribute__((ext_vector_type(2)));
typedef float v8f __attribute__((ext_vector_type(8)));
typedef unsigned int v4u __attribute__((ext_vector_type(4)));
typedef int v4i __attribute__((ext_vector_type(4)));
typedef int v8i __attribute__((ext_vector_type(8)));

// ---------------- workspace layout (floats) ----------------
// ws[0]              : C_SIG
// ws[4 .. 4+169)     : D_PRIM (13x13 row-major)
// ws[176 .. 176+614) : W3J tensors, per-key flattened (i*dj+j)*dk+k
#define DPB   4
#define W3JB  176

// Active-instruction metadata (INSTR order == w2 column order, 22 entries).
__constant__ int Q_GOFF[22] = {0,1,2,9,24,33,82, 3,6,18,21,52,67, 13,28,37,42,47,86, 60,75, 98};
__constant__ int Q_PA [22] = {0,1,1,2,5,5,8, 2,2,5,5,8,8, 2,5,5,8,8,8, 8,8, 8};
__constant__ int Q_PB [22] = {0,0,1,2,2,5,8, 0,1,0,1,2,5, 2,2,5,0,1,8, 2,5, 8};
__constant__ int Q_DI [22] = {1,1,1,3,3,3,5, 3,3,3,3,5,5, 3,3,3,5,5,5, 5,5, 5};
__constant__ int Q_DJ [22] = {1,1,1,3,3,3,5, 1,1,1,1,3,3, 3,3,3,1,1,5, 3,3, 5};
__constant__ int Q_DK [22] = {1,1,1,1,1,1,1, 3,3,3,3,3,3, 5,5,5,5,5,5, 7,7, 9};
__constant__ int Q_W3 [22] = {0,0,0,1,1,1,10, 35,35,35,35,44,44, 89,89,89,134,134,159, 284,284, 389};
__constant__ int Q_C  [22] = {0,0,0,0,0,0,0, 1,1,1,1,1,1, 2,2,2,2,2,2, 3,3, 4};

// Real spherical harmonics (reference normalization), X=v[2], Y=v[0], Z=v[1].
__device__ inline void sh_all(float X, float Y, float Z, float* o) {
    float X2 = X*X, Y2 = Y*Y, Z2 = Z*Z;
    o[0]  = 1.0f;
    o[1]  = 1.7320508f*Y;  o[2] = 1.7320508f*Z;  o[3] = 1.7320508f*X;
    o[4]  = 3.8729833f*X*Y;
    o[5]  = 3.8729833f*Y*Z;
    o[6]  = 1.1180340f*(3.f*Z2-1.f);
    o[7]  = 3.8729833f*X*Z;
    o[8]  = 1.9364917f*(X2-Y2);
    o[9]  = 2.0916501f*Y*(3.f*X2-Y2);
    o[10] = 10.2469508f*X*Y*Z;
    o[11] = 1.6201852f*Y*(5.f*Z2-1.f);
    o[12] = 1.3228757f*(5.f*Z2*Z-3.f*Z);
    o[13] = 1.6201852f*X*(5.f*Z2-1.f);
    o[14] = 5.1234754f*Z*(X2-Y2);
    o[15] = 2.0916501f*X*(X2-3.f*Y2);
    o[16] = 8.8741197f*X*Y*(X2-Y2);
    o[17] = 6.2749502f*Y*Z*(3.f*X2-Y2);
    o[18] = 3.3541020f*X*Y*(7.f*Z2-1.f);
    o[19] = 2.3717082f*Y*Z*(7.f*Z2-3.f);
    o[20] = 0.375f*(35.f*Z2*Z2-30.f*Z2+3.f);
    o[21] = 2.3717082f*X*Z*(7.f*Z2-3.f);
    o[22] = 1.6770510f*(X2-Y2)*(7.f*Z2-1.f);
    o[23] = 6.2749502f*X*Z*(X2-3.f*Y2);
    o[24] = 2.2185299f*(X2*X2-6.f*X2*Y2+Y2*Y2);
}

// Tensor Data Mover: 1-row tile of nElems f32 from global -> LDS.
// D# built per CDNA5 ISA 8.3/8.4: group0 {count=1, lds_addr, global_addr(57b), type=2},
// group1 {mask=0, data_size=4B, tensor_dim0=tile_dim0=stride=nElems, dims1/2=1,0}.
__device__ inline void tdm_load_1d(const void* gptr, unsigned ldsOff, unsigned nElems) {
    unsigned long long ga = (unsigned long long)(uintptr_t)gptr;
    v4u g0;
    g0[0] = 1u;                                          // count=1, user mode
    g0[1] = ldsOff;                                      // lds_addr (bytes)
    g0[2] = (unsigned)(ga & 0xFFFFFFFFu);                // global_addr[31:0]
    g0[3] = (unsigned)((ga >> 32) & 0x1FFFFFFu) | (2u << 30); // global_addr[56:32] | type=2
    v8i g1;
    g1[0] = (int)(2u << 16);                             // workgroup_mask=0, data_size=4B
    g1[1] = (int)((nElems & 0xFFFFu) << 16);             // tensor_dim0[15:0]
    g1[2] = (int)(1u << 16);                             // tensor_dim0 hi=0 | tensor_dim1=1
    g1[3] = (int)((nElems & 0xFFFFu) << 16);             // tensor_dim1 hi=0 | tile_dim0
    g1[4] = 1;                                           // tile_dim1=1, tile_dim2=0
    g1[5] = (int)nElems;                                 // tensor_dim0_stride lo
    g1[6] = 0;                                           // stride hi / dim1_stride lo
    g1[7] = 0;
    v4i z4 = {0,0,0,0};
#if defined(__clang_major__) && (__clang_major__ >= 23)
    v8i z8 = {0,0,0,0,0,0,0,0};
    __builtin_amdgcn_tensor_load_to_lds(g0, g1, z4, z4, z8, 0);
#else
    __builtin_amdgcn_tensor_load_to_lds(g0, g1, z4, z4, 0);
#endif
}

// -------- precompute W3J (exact Gauss-Legendre x azimuthal quadrature),
//          D_PRIM, and C_SIG into the workspace. One wave, trivially cheap. --------
__global__ void pao_precompute(float* __restrict__ ws) {
    const int tid = threadIdx.x;
    const float glz[5] = {-0.9061798459f,-0.5384693101f,0.f,0.5384693101f,0.9061798459f};
    const float glw[5] = { 0.2369268851f, 0.4786286705f,0.5688888889f,0.4786286705f,0.2369268851f};
    if (tid < 10) {
        const int L1[10]={0,1,2,1,2,1,2,2,2,2};
        const int L2[10]={0,1,2,0,1,1,0,2,1,2};
        const int L3[10]={0,0,0,1,1,2,2,2,3,4};
        const int OFFK[10]={0,1,10,35,44,89,134,159,284,389};
        const int l1=L1[tid], l2=L2[tid], l3=L3[tid];
        const int d1=2*l1+1, d2=2*l2+1, d3=2*l3+1;
        const int ne = d1*d2*d3;
        float C[225];
        for (int e=0;e<ne;++e) C[e]=0.f;
        for (int iq=0;iq<5;++iq) {
            float z = glz[iq]; float st = sqrtf(fmaxf(0.f,1.f-z*z));
            float wgt = glw[iq]*(0.5f/16.f);
            for (int ip=0;ip<16;++ip) {
                float ph = 6.28318530718f*(float)ip/16.f;
                float v0 = st*cosf(ph), v1 = st*sinf(ph), v2 = z;
                float sh[25]; sh_all(v2, v0, v1, sh);
                const float* Y1 = sh + l1*l1;
                const float* Y2 = sh + l2*l2;
                const float* Y3 = sh + l3*l3;
                int e=0;
                for (int i=0;i<d1;++i) for (int j=0;j<d2;++j) {
                    float pij = wgt*Y1[i]*Y2[j];
                    for (int k=0;k<d3;++k) C[e++] += pij*Y3[k];
                }
            }
        }
        float nrm2 = 0.f; float mx = -1.f; int mi = 0;
        for (int e=0;e<ne;++e) { nrm2 += C[e]*C[e];
            float a = fabsf(C[e]); if (a > mx) { mx = a; mi = e; } }
        float scale = ((C[mi] < 0.f) ? -1.f : 1.f) / sqrtf(nrm2);
        for (int e=0;e<ne;++e) ws[W3JB + OFFK[tid] + e] = C[e]*scale;
    } else if (tid == 10) {
        // D_PRIM: block-diag Wigner-D for R_YZX (Rv = (v2, v0, v1)); exact quadrature.
        float D1[9], D2[25];
        for (int e=0;e<9;++e)  D1[e]=0.f;
        for (int e=0;e<25;++e) D2[e]=0.f;
        for (int iq=0;iq<5;++iq) {
            float z = glz[iq]; float st = sqrtf(fmaxf(0.f,1.f-z*z));
            float wgt = glw[iq]*(0.5f/16.f);
            for (int ip=0;ip<16;++ip) {
                float ph = 6.28318530718f*(float)ip/16.f;
                float v0 = st*cosf(ph), v1 = st*sinf(ph), v2 = z;
                float shn[25]; sh_all(v2, v0, v1, shn);       // Y(n)
                float shr[25]; sh_all(v1, v2, v0, shr);       // Y(Rn)
                for (int i=0;i<3;++i) for (int j=0;j<3;++j)
                    D1[i*3+j] += wgt*shr[1+i]*shn[1+j];
                for (int i=0;i<5;++i) for (int j=0;j<5;++j)
                    D2[i*5+j] += wgt*shr[4+i]*shn[4+j];
            }
        }
        for (int e=0;e<169;++e) ws[DPB+e] = 0.f;
        ws[DPB + 0*13 + 0] = 1.f;
        ws[DPB + 1*13 + 1] = 1.f;
        for (int i=0;i<3;++i) for (int j=0;j<3;++j) {
            ws[DPB + (2+i)*13 + (2+j)] = D1[i*3+j];
            ws[DPB + (5+i)*13 + (5+j)] = D1[i*3+j];
        }
        for (int i=0;i<5;++i) for (int j=0;j<5;++j)
            ws[DPB + (8+i)*13 + (8+j)] = D2[i*5+j];
    } else if (tid == 11) {
        // C_SIG = 1/sqrt(E[sigmoid(z)^2]), z~N(0,1), same grid as reference.
        float m2 = 0.f; const float dz = 24.f/40000.f;
        for (int i=0;i<=40000;++i) {
            float z = -12.f + dz*(float)i;
            float s = 1.f/(1.f+expf(-z));
            float p = expf(-0.5f*z*z)*0.39894228f;
            m2 += s*s*p;
        }
        m2 *= dz;
        ws[0] = 1.f/sqrtf(m2);
    }
}

// -------- main kernel: 1 wave / block, 32 items / block --------
__global__ void __launch_bounds__(32)
pao_forward(const float* __restrict__ nbr, const float* __restrict__ feat,
            const float* __restrict__ w1g, const float* __restrict__ w2g,
            const float* __restrict__ ws, float* __restrict__ out, int nItems)
{
    __shared__ float embAll[192][12];  // rows = item-in-block*6 + neighbor, K padded to 12
    __shared__ float hT[16][17];       // per-tile hidden activations (padded: no bank conflicts)
    __shared__ float wAll[192][49];    // MLP outputs (44 cols, padded stride 49)
    __shared__ float w1L[12][16];
    __shared__ float w2L[16][48];
    __shared__ float tdmW1[160];       // TDM landing buffers (raw weights)
    __shared__ float tdmW2[704];

    const int lane = threadIdx.x;
    const int bRaw = blockIdx.x*32 + lane;
    const int b = (bRaw < nItems) ? bRaw : (nItems-1);

    __builtin_prefetch((const void*)nbr, 0, 1);   // global_prefetch_b8

    // Async DMA of both weight matrices into LDS via the Tensor Data Mover;
    // overlaps with per-lane embedding computation below.
    tdm_load_1d((const void*)w1g, (unsigned)(uintptr_t)(void*)tdmW1, 160u);
    tdm_load_1d((const void*)w2g, (unsigned)(uintptr_t)(void*)tdmW2, 704u);

    float nb[6][3], ft[6][2];
    #pragma unroll
    for (int n=0;n<6;++n) {
        nb[n][0]=nbr[b*18+n*3+0]; nb[n][1]=nbr[b*18+n*3+1]; nb[n][2]=nbr[b*18+n*3+2];
        ft[n][0]=feat[b*12+n*2+0]; ft[n][1]=feat[b*12+n*2+1];
    }

    // Phase 1: radial bump embeddings -> LDS rows
    #pragma unroll
    for (int n=0;n<6;++n) {
        float r = sqrtf(nb[n][0]*nb[n][0]+nb[n][1]*nb[n][1]+nb[n][2]*nb[n][2]);
        float rd = r*(11.f/6.f);
        int row = lane*6+n;
        #pragma unroll
        for (int i=0;i<10;++i) {
            float d = rd - (float)(i+1);
            float t1 = 1.f+d, t2 = 1.f-d;
            float e = 0.f;
            if (t1 > 0.f && t2 > 0.f)
                e = 26.66930f*__expf(-1.f/t1 - 1.f/t2);  // 1.14136*e^2*sqrt(10)*sus*sus
            embAll[row][i] = e;
        }
        embAll[row][10]=0.f; embAll[row][11]=0.f;
    }

    // Wait for TDM, then build zero-padded weight tiles from the LDS copies.
    __builtin_amdgcn_s_wait_tensorcnt(0);
    __syncthreads();
    for (int idx = lane; idx < 12*16; idx += 32) {
        int k = idx >> 4, n = idx & 15;
        w1L[k][n] = (k < 10) ? tdmW1[k*16+n] : 0.f;
    }
    for (int idx = lane; idx < 16*48; idx += 32) {
        int k = idx / 48, n = idx % 48;
        w2L[k][n] = (n < 44) ? tdmW2[k*44+n] : 0.f;
    }
    __syncthreads();

    const float cSig  = ws[0];
    const int mrow  = lane & 15;          // M-row / N-col owned by this lane
    const int khalf = (lane >> 4) << 1;   // K pair offset (0 or 2) per WMMA f32 layout
    const int mbase = (lane >> 4) << 3;   // C/D: M = v (+8 for upper half-wave)

    // Phase 2: 12 M-tiles of 16 rows through GEMM1(+sigmoid) and GEMM2 via WMMA f32 16x16x4
    for (int t = 0; t < 12; ++t) {
        v8f c1 = {};
        #pragma unroll
        for (int s = 0; s < 3; ++s) {
            int k0 = 4*s;
            v2f a, bf;
            a[0]  = embAll[t*16+mrow][k0+khalf];
            a[1]  = embAll[t*16+mrow][k0+khalf+1];
            bf[0] = w1L[k0+khalf][mrow];
            bf[1] = w1L[k0+khalf+1][mrow];
            c1 = __builtin_amdgcn_wmma_f32_16x16x4_f32(false, a, false, bf, (short)0, c1, false, false);
        }
        #pragma unroll
        for (int v=0;v<8;++v) {
            // sigmoid(acc/sqrt(10))*C_SIG with fast hardware reciprocal (v_rcp_f32)
            float e = __expf(-c1[v]*0.31622777f);
            hT[v+mbase][mrow] = cSig * __builtin_amdgcn_rcpf(1.f + e);
        }
        __syncthreads();
        #pragma unroll
        for (int nt=0; nt<3; ++nt) {
            v8f c2 = {};
            #pragma unroll
            for (int s=0;s<4;++s) {
                int k0 = 4*s;
                v2f a2, b2;
                a2[0] = hT[mrow][k0+khalf];
                a2[1] = hT[mrow][k0+khalf+1];
                b2[0] = w2L[k0+khalf][nt*16+mrow];
                b2[1] = w2L[k0+khalf+1][nt*16+mrow];
                c2 = __builtin_amdgcn_wmma_f32_16x16x4_f32(false, a2, false, b2, (short)0, c2, false, false);
            }
            #pragma unroll
            for (int v=0;v<8;++v)
                wAll[t*16+v+mbase][nt*16+mrow] = c2[v]*0.25f;  // /sqrt(16)
        }
        __syncthreads();
    }

    // Phase 3: per-lane (per-item) equivariant vector h_aux[107]
    float haux[107];
    #pragma unroll
    for (int i=0;i<107;++i) haux[i]=0.f;
    for (int n=0;n<6;++n) {
        float r = sqrtf(nb[n][0]*nb[n][0]+nb[n][1]*nb[n][1]+nb[n][2]*nb[n][2]);
        float inv = 1.f/r;
        float shv[25];
        sh_all(nb[n][2]*inv, nb[n][0]*inv, nb[n][1]*inv, shv);
        int row = lane*6+n;
        float f0 = ft[n][0], f1 = ft[n][1];
        #pragma unroll
        for (int q=0;q<22;++q) {
            float s = (wAll[row][2*q]*f0 + wAll[row][2*q+1]*f1)*0.70710678f; // /sqrt(2)
            int c = Q_C[q]; int goff = Q_GOFF[q]; int so = c*c; int dk = 2*c+1;
            for (int k=0;k<dk;++k) haux[goff+k] += s*shv[so+k];
        }
    }
    #pragma unroll
    for (int i=0;i<107;++i) haux[i] *= 0.40824829f;  // /sqrt(6)

    // Phase 4: assemble M = I + sum_g W3J-blocks, symmetrize (tril + tril^T)
    float Mm[13][13];
    for (int r=0;r<13;++r) for (int c=0;c<13;++c) Mm[r][c] = (r==c)?1.f:0.f;
    #pragma unroll
    for (int q=0;q<22;++q) {
        int di=Q_DI[q], dj=Q_DJ[q], dk=Q_DK[q];
        int pa=Q_PA[q], pb=Q_PB[q], goff=Q_GOFF[q];
        const float* W = ws + W3JB + Q_W3[q];
        for (int i=0;i<di;++i) for (int j=0;j<dj;++j) {
            const float* Wij = W + (i*dj+j)*dk;
            float acc = 0.f;
            for (int k=0;k<dk;++k) acc += haux[goff+k]*Wij[k];
            Mm[pa+i][pb+j] += acc;
        }
    }
    float Am[13][13];
    for (int r=0;r<13;++r) for (int c=0;c<13;++c)
        Am[r][c] = (r>=c) ? Mm[r][c] : Mm[c][r];

    // Phase 5: cyclic Jacobi eigensolve (13x13 symmetric), per lane
    float Vm[13][13];
    for (int r=0;r<13;++r) for (int c=0;c<13;++c) Vm[r][c] = (r==c)?1.f:0.f;
    for (int sweep=0; sweep<12; ++sweep) {
        for (int p=0;p<12;++p) for (int q2=p+1;q2<13;++q2) {
            float apq = Am[p][q2];
            if (fabsf(apq) <= 1e-10f) continue;
            float theta = (Am[q2][q2]-Am[p][p])/(2.f*apq);
            float tt = (theta >= 0.f) ? 1.f/(theta+sqrtf(theta*theta+1.f))
                                      : 1.f/(theta-sqrtf(theta*theta+1.f));
            float cc = 1.f/sqrtf(tt*tt+1.f);
            float ssn = tt*cc;
            for (int k=0;k<13;++k) { float a1=Am[k][p], a2=Am[k][q2];
                Am[k][p]=cc*a1-ssn*a2; Am[k][q2]=ssn*a1+cc*a2; }
            for (int k=0;k<13;++k) { float a1=Am[p][k], a2=Am[q2][k];
                Am[p][k]=cc*a1-ssn*a2; Am[q2][k]=ssn*a1+cc*a2; }
            for (int k=0;k<13;++k) { float v1=Vm[k][p], v2e=Vm[k][q2];
                Vm[k][p]=cc*v1-ssn*v2e; Vm[k][q2]=ssn*v1+cc*v2e; }
        }
    }

    // Phase 6: 4 smallest eigenvalues (ascending), project eigenvectors through D_PRIM
    if (bRaw < nItems) {
        bool taken[13];
        for (int i=0;i<13;++i) taken[i]=false;
        for (int sel=0; sel<4; ++sel) {
            int best = 0; float bv = 3.0e38f;
            for (int p=0;p<13;++p)
                if (!taken[p] && Am[p][p] < bv) { bv = Am[p][p]; best = p; }
            taken[best] = true;
            for (int cc2=0; cc2<13; ++cc2) {
                float acc = 0.f;
                for (int r=0;r<13;++r) acc += Vm[r][best]*ws[DPB + r*13 + cc2];
                out[bRaw*52 + sel*13 + cc2] = acc;
            }
        }
    }
}

extern "C" void kernel_launch(void* const* d_in, const int* in_sizes, int n_in,
                              void* d_out, int out_size, void* d_ws, size_t ws_size,
                              hipStream_t stream) {
    const float* neighbors = (const float*)d_in[0];
    const float* features  = (const float*)d_in[1];
    const float* w1        = (const float*)d_in[2];
    const float* w2        = (const float*)d_in[3];
    float* out = (float*)d_out;
    float* ws  = (float*)d_ws;
    int nItems = in_sizes[0] / 18;   // (B, 6, 3)

    pao_precompute<<<1, 32, 0, stream>>>(ws);
    int nBlocks = (nItems + 31) / 32;
    pao_forward<<<nBlocks, 32, 0, stream>>>(neighbors, features, w1, w2, ws, out, nItems);

    (void)n_in; (void)out_size; (void)ws_size;
}